// myZCANorm_38560216383953
// MI455X (gfx1250) — compile-verified
//
#include <hip/hip_runtime.h>
#include <hip/hip_bf16.h>

// ---------------------------------------------------------------------------
// ZCA whitening for x:(64,64,64,64) on MI455X (gfx1250, wave32).
// Bandwidth-bound (~256MB traffic @ 23.3TB/s ~ 11us). GEMMs run on
// V_WMMA_F32_16X16X4_F32; all bulk data movement uses the CDNA5 async
// Global<->LDS engine (GLOBAL_LOAD_ASYNC_TO_LDS_B128 /
// GLOBAL_STORE_ASYNC_FROM_LDS_B128, ASYNCcnt-tracked) through 320KB WGP LDS.
// ---------------------------------------------------------------------------

typedef __attribute__((ext_vector_type(2))) float v2f;
typedef __attribute__((ext_vector_type(8))) float v8f;

#define C_CH      64
#define HW_       4096
#define NSLAB     64
#define M_TOT     262144.0        // N*H*W samples per channel
#define EPSV      1e-5
#define ETHRESH   0.95
#define CHUNK     512             // columns of x staged per block
#define XS_STRIDE 516             // 516%64=4 -> conflict-free across rows; 516*4%16==0
#define AF_STRIDE 68              // 68%64=4

// Workspace byte offsets
#define WS_S      0               // float[64*64]   raw Gram sums  (16384 B)
#define WS_MU     16384           // double[64]     channel means  (512 B)
#define WS_RSTD   16896           // double[64]     1/sqrt(var+eps)(512 B)
#define WS_AFIN   17408           // float[64*64]   fused output matrix (16384 B)
#define WS_BFIN   33792           // float[64]      fused output bias   (256 B)

// ---------------------------------------------------------------------------
// CDNA5 async Global<->LDS helpers (ISA 15.18, opcodes 95-102; ASYNCcnt).
// LDS operand is a per-lane byte offset: for AMDGPU the low 32 bits of a
// generic pointer into __shared__ are exactly the addrspace(3) offset.
// ---------------------------------------------------------------------------
__device__ __forceinline__ unsigned lds_off32(const void* p) {
  return (unsigned)(unsigned long long)(uintptr_t)p;
}
__device__ __forceinline__ void async_load_b128(unsigned ldsoff, const float* g) {
  asm volatile("global_load_async_to_lds_b128 %0, %1, off"
               :: "v"(ldsoff), "v"(g) : "memory");
}
__device__ __forceinline__ void async_store_b128(float* g, unsigned ldsoff) {
  asm volatile("global_store_async_from_lds_b128 %0, %1, off"
               :: "v"(g), "v"(ldsoff) : "memory");
}
__device__ __forceinline__ void wait_async0() {
  asm volatile("s_wait_asynccnt 0x0" ::: "memory");
}

// ---------------------------------------------------------------------------
// Kernel 1: per-channel mean / unbiased var (f64 accumulate). Also zeroes S.
// ---------------------------------------------------------------------------
__global__ void zca_stats(const float* __restrict__ x, float* __restrict__ S,
                          double* __restrict__ mu, double* __restrict__ rstd) {
  __shared__ double rs[256], rs2[256];
  const int c = blockIdx.x, tid = threadIdx.x;
  if (tid < 64) S[c * 64 + tid] = 0.0f;           // init Gram accumulator row
  double s = 0.0, s2 = 0.0;
  for (int n = 0; n < NSLAB; ++n) {
    const float* p = x + ((size_t)n * C_CH + c) * HW_;
    for (int i = tid; i < HW_; i += 256) {
      double v = (double)p[i];
      s += v; s2 += v * v;
    }
  }
  rs[tid] = s; rs2[tid] = s2;
  __syncthreads();
  for (int off = 128; off > 0; off >>= 1) {
    if (tid < off) { rs[tid] += rs[tid + off]; rs2[tid] += rs2[tid + off]; }
    __syncthreads();
  }
  if (tid == 0) {
    double m   = rs[0] / M_TOT;
    double var = (rs2[0] - rs[0] * rs[0] / M_TOT) / (M_TOT - 1.0);
    mu[c]   = m;
    rstd[c] = 1.0 / sqrt(var + EPSV);
  }
}

// ---------------------------------------------------------------------------
// Kernel 2: raw Gram S[i][j] = sum_m x_i[m]*x_j[m] via fp32 WMMA.
// 512 blocks; each async-copies a 64x512 chunk of x^(C,M) into LDS, 8 waves
// cover the 4x4 grid of 16x16 tiles (2 tiles/wave sharing the A fragment),
// flush partials with global_atomic_add_f32.
// ---------------------------------------------------------------------------
__global__ void zca_gram(const float* __restrict__ x, float* __restrict__ S) {
  extern __shared__ float xs[];                   // [64][XS_STRIDE]
  const int tid   = threadIdx.x;
  const int slab  = blockIdx.x >> 3;              // n
  const int chunk = blockIdx.x & 7;               // 512-col chunk in slab
  const float* base = x + (size_t)slab * C_CH * HW_ + chunk * CHUNK;

  for (int i = tid; i < C_CH * (CHUNK / 4); i += 256) {  // 8192 x 16B copies
    int row = i >> 7, c4 = (i & 127) << 2;
    async_load_b128(lds_off32(&xs[row * XS_STRIDE + c4]),
                    base + (size_t)row * HW_ + c4);
  }
  wait_async0();
  __syncthreads();

  const int wave = tid >> 5, lane = tid & 31;
  const int rsel = lane & 15;                     // A/B fragment row select
  const int ksel = (lane >> 4) << 1;              // K 0/1 (lanes<16) or 2/3
  const int t0  = 2 * wave;
  const int i0  = (t0 >> 2) * 16;                 // same for t0 and t0+1
  const int j0a = (t0 & 3) * 16;
  const int j0b = ((t0 + 1) & 3) * 16;

  v8f acc0 = {}, acc1 = {};
  for (int kk = 0; kk < CHUNK; kk += 4) {
    v2f a  = *(const v2f*)&xs[(i0  + rsel) * XS_STRIDE + kk + ksel];
    v2f b0 = *(const v2f*)&xs[(j0a + rsel) * XS_STRIDE + kk + ksel];
    v2f b1 = *(const v2f*)&xs[(j0b + rsel) * XS_STRIDE + kk + ksel];
    acc0 = __builtin_amdgcn_wmma_f32_16x16x4_f32(false, a, false, b0,
                                                 (short)0, acc0, false, false);
    acc1 = __builtin_amdgcn_wmma_f32_16x16x4_f32(false, a, false, b1,
                                                 (short)0, acc1, false, false);
  }
  const int mhi = (lane >> 4) * 8, ncol = lane & 15;
  for (int r = 0; r < 8; ++r) {                   // C/D: row = r + 8*(lane>=16)
    atomicAdd(&S[(i0 + r + mhi) * 64 + (j0a + ncol)], acc0[r]);
    atomicAdd(&S[(i0 + r + mhi) * 64 + (j0b + ncol)], acc1[r]);
  }
}

// ---------------------------------------------------------------------------
// Kernel 3: 64x64 f64 eigen path (single block, 64 lanes; row per thread).
// M0 from raw S + stats; power iteration + deflation until 95% energy
// (sum of eigenvalues == trace(M0)); fuse weight/rstd/mu/bias into Afin/bfin.
// ---------------------------------------------------------------------------
__global__ void zca_solve(const float* __restrict__ S,
                          const double* __restrict__ mu,
                          const double* __restrict__ rstd,
                          const float* __restrict__ weight,
                          const float* __restrict__ bias,
                          float* __restrict__ Afin, float* __restrict__ bfin) {
  extern __shared__ double dsm[];
  double* Md  = dsm;                // [64*65] padded
  double* Wx  = Md + 64 * 65;       // [64*64]
  double* v   = Wx + 64 * 64;       // [64]
  double* mv  = v + 64;             // [64]
  double* red = mv + 64;            // [64]
  __shared__ double scl[2];         // [0]=sum_e (trace), [1]=cum energy
  const int r = threadIdx.x;        // 0..63

  const double rst_r = rstd[r], mu_r = mu[r];
  for (int c = 0; c < 64; ++c) {
    double val = rst_r * rstd[c] *
                 ((double)S[r * 64 + c] - M_TOT * mu_r * mu[c]) / M_TOT;
    if (c == r) val += EPSV;
    Md[r * 65 + c] = val;
    Wx[r * 64 + c] = 0.0;
  }
  __syncthreads();
  red[r] = Md[r * 65 + r];          // trace = sum of eigenvalues
  __syncthreads();
  for (int off = 32; off > 0; off >>= 1) {
    if (r < off) red[r] += red[r + off];
    __syncthreads();
  }
  if (r == 0) { scl[0] = red[0]; scl[1] = 0.0; }
  __syncthreads();
  const double sum_e = scl[0];

  for (int comp = 0; comp < 32; ++comp) {
    v[r] = Md[r * 65 + comp] + ((r == comp) ? 1e-6 : 0.0);
    __syncthreads();
    for (int it = 0; it < 64; ++it) {           // power iteration
      double acc = 0.0;
      for (int c = 0; c < 64; ++c) acc += Md[r * 65 + c] * v[c];
      mv[r] = acc;
      __syncthreads();
      red[r] = acc * acc;
      __syncthreads();
      for (int off = 32; off > 0; off >>= 1) {
        if (r < off) red[r] += red[r + off];
        __syncthreads();
      }
      double inv = 1.0 / sqrt(red[0] + 1e-300);
      __syncthreads();
      v[r] = mv[r] * inv;
      __syncthreads();
    }
    double acc = 0.0;                           // mv = M v ; lam = v.Mv
    for (int c = 0; c < 64; ++c) acc += Md[r * 65 + c] * v[c];
    mv[r] = acc;
    __syncthreads();
    red[r] = v[r] * mv[r];
    __syncthreads();
    for (int off = 32; off > 0; off >>= 1) {
      if (r < off) red[r] += red[r + off];
      __syncthreads();
    }
    const double lam = red[0];
    __syncthreads();
    const double s = 1.0 / sqrt(lam), vr = v[r];
    for (int c = 0; c < 64; ++c) Wx[r * 64 + c] += vr * v[c] * s;
    if (r == 0) scl[1] += lam;
    __syncthreads();
    if ((scl[1] / sum_e) >= ETHRESH) break;     // uniform across block
    const double mvr = mv[r];
    for (int c = 0; c < 64; ++c) Md[r * 65 + c] -= mvr * v[c];
    __syncthreads();
  }
  __syncthreads();
  // out = (Wxr@xh)*w + b  ==  Afin@x_raw + bfin, with
  //   Afin[co][c] = w[co]*Wxr[co][c]*rstd[c],  bfin[co] = b[co] - Afin[co].mu
  const double w_r = (double)weight[r], b_r = (double)bias[r];
  double dot = 0.0;
  for (int c = 0; c < 64; ++c) {
    double a = w_r * Wx[r * 64 + c] * rstd[c];
    Afin[r * 64 + c] = (float)a;
    dot += a * mu[c];
  }
  bfin[r] = (float)(b_r - dot);
}

// ---------------------------------------------------------------------------
// Kernel 4: out(C,M tile) = Afin(64x64) @ x_raw + bfin via fp32 WMMA.
// Async-load x/Afin/bfin into LDS; each wave: 4 m-tiles x 4 co-tiles;
// results scattered to an LDS out-stage and drained with coalesced
// GLOBAL_STORE_ASYNC_FROM_LDS_B128. Total LDS ~275KB of the 320KB WGP pool.
// ---------------------------------------------------------------------------
__global__ void zca_out(const float* __restrict__ x,
                        const float* __restrict__ Afin,
                        const float* __restrict__ bfin,
                        float* __restrict__ out) {
  extern __shared__ float smem[];
  float* xs   = smem;                             // [64][XS_STRIDE]  input stage
  float* af   = smem + 64 * XS_STRIDE;            // [64][AF_STRIDE]
  float* bf   = af + 64 * AF_STRIDE;              // [64]
  float* outb = bf + 64;                          // [64][XS_STRIDE]  output stage
  const int tid   = threadIdx.x;
  const int slab  = blockIdx.x >> 3;
  const int chunk = blockIdx.x & 7;
  const float* base = x + (size_t)slab * C_CH * HW_ + chunk * CHUNK;

  for (int i = tid; i < C_CH * (CHUNK / 4); i += 256) {
    int row = i >> 7, c4 = (i & 127) << 2;
    async_load_b128(lds_off32(&xs[row * XS_STRIDE + c4]),
                    base + (size_t)row * HW_ + c4);
  }
  for (int i = tid; i < C_CH * 16; i += 256) {    // 64x64 Afin
    int row = i >> 4, c4 = (i & 15) << 2;
    async_load_b128(lds_off32(&af[row * AF_STRIDE + c4]),
                    Afin + row * 64 + c4);
  }
  if (tid < 16) async_load_b128(lds_off32(&bf[tid * 4]), bfin + tid * 4);
  wait_async0();
  __syncthreads();

  const int wave = tid >> 5, lane = tid & 31;
  const int rsel = lane & 15;
  const int ksel = (lane >> 4) << 1;
  const int mhi  = (lane >> 4) * 8;

  for (int q = 0; q < 4; ++q) {
    const int mt   = wave * 4 + q;                // m-tile 0..31
    const int mloc = mt * 16 + rsel;              // B fragment column (N)
    v8f acc0 = {}, acc1 = {}, acc2 = {}, acc3 = {};
    for (int kk = 0; kk < 64; kk += 4) {          // K = input channels
      v2f b;
      b.x = xs[(kk + ksel)     * XS_STRIDE + mloc];
      b.y = xs[(kk + ksel + 1) * XS_STRIDE + mloc];
      v2f a0 = *(const v2f*)&af[( 0 + rsel) * AF_STRIDE + kk + ksel];
      v2f a1 = *(const v2f*)&af[(16 + rsel) * AF_STRIDE + kk + ksel];
      v2f a2 = *(const v2f*)&af[(32 + rsel) * AF_STRIDE + kk + ksel];
      v2f a3 = *(const v2f*)&af[(48 + rsel) * AF_STRIDE + kk + ksel];
      acc0 = __builtin_amdgcn_wmma_f32_16x16x4_f32(false, a0, false, b,
                                                   (short)0, acc0, false, false);
      acc1 = __builtin_amdgcn_wmma_f32_16x16x4_f32(false, a1, false, b,
                                                   (short)0, acc1, false, false);
      acc2 = __builtin_amdgcn_wmma_f32_16x16x4_f32(false, a2, false, b,
                                                   (short)0, acc2, false, false);
      acc3 = __builtin_amdgcn_wmma_f32_16x16x4_f32(false, a3, false, b,
                                                   (short)0, acc3, false, false);
    }
    const int mcol = mt * 16 + (lane & 15);       // D: row = r + 8*(lane>=16)
    for (int r = 0; r < 8; ++r) {
      const int co = r + mhi;
      outb[(co)      * XS_STRIDE + mcol] = acc0[r] + bf[co];
      outb[(co + 16) * XS_STRIDE + mcol] = acc1[r] + bf[co + 16];
      outb[(co + 32) * XS_STRIDE + mcol] = acc2[r] + bf[co + 32];
      outb[(co + 48) * XS_STRIDE + mcol] = acc3[r] + bf[co + 48];
    }
  }
  __syncthreads();                                // out-stage fully populated

  float* outbase = out + (size_t)slab * C_CH * HW_ + chunk * CHUNK;
  for (int i = tid; i < C_CH * (CHUNK / 4); i += 256) {
    int row = i >> 7, c4 = (i & 127) << 2;
    async_store_b128(outbase + (size_t)row * HW_ + c4,
                     lds_off32(&outb[row * XS_STRIDE + c4]));
  }
  wait_async0();                                  // (S_ENDPGM also waits idle)
}

// ---------------------------------------------------------------------------
extern "C" void kernel_launch(void* const* d_in, const int* in_sizes, int n_in,
                              void* d_out, int out_size, void* d_ws, size_t ws_size,
                              hipStream_t stream) {
  (void)in_sizes; (void)n_in; (void)out_size; (void)ws_size;
  const float* x      = (const float*)d_in[0];
  const float* weight = (const float*)d_in[1];
  const float* bias   = (const float*)d_in[2];
  float* out = (float*)d_out;
  char* ws = (char*)d_ws;
  float*  S    = (float*)(ws + WS_S);
  double* mu   = (double*)(ws + WS_MU);
  double* rstd = (double*)(ws + WS_RSTD);
  float*  Afin = (float*)(ws + WS_AFIN);
  float*  bfin = (float*)(ws + WS_BFIN);

  const size_t xs_bytes    = (size_t)64 * XS_STRIDE * sizeof(float);       // 129KB
  const size_t out_bytes   = xs_bytes                                       // x stage
                           + (size_t)64 * AF_STRIDE * sizeof(float)         // Afin
                           + 64 * sizeof(float)                             // bfin
                           + xs_bytes;                                      // out stage (~275KB)
  const size_t solve_bytes = (size_t)(64 * 65 + 64 * 64 + 3 * 64) * sizeof(double);

  zca_stats<<<64, 256, 0, stream>>>(x, S, mu, rstd);
  zca_gram <<<512, 256, xs_bytes, stream>>>(x, S);
  zca_solve<<<1, 64, solve_bytes, stream>>>(S, mu, rstd, weight, bias, Afin, bfin);
  zca_out  <<<512, 256, out_bytes, stream>>>(x, Afin, bfin, out);
}